// ST_HGAT_16140487098676
// MI455X (gfx1250) — compile-verified
//
#include <hip/hip_runtime.h>
#include <hip/hip_bf16.h>
#include <math.h>

// ---------------------------------------------------------------------------
// ST_HGAT forward on gfx1250 (CDNA5).  f32 end-to-end; node linears use
// V_WMMA_F32_16X16X4_F32 (exact fp32 matrix op), edge phase is wave32-
// parallel gather / segment-softmax / atomic scatter.
// ---------------------------------------------------------------------------

#define AH 128   // H*D
#define NH 4     // heads
#define HD 32    // dim per head (== wave32 width)

typedef float v2f __attribute__((ext_vector_type(2)));
typedef float v8f __attribute__((ext_vector_type(8)));

// ---- ordered-uint encoding for float atomic max ---------------------------
__device__ __forceinline__ unsigned fenc(float f) {
    unsigned u = __float_as_uint(f);
    return (u & 0x80000000u) ? ~u : (u | 0x80000000u);
}
__device__ __forceinline__ float fdec(unsigned u) {
    return (u & 0x80000000u) ? __uint_as_float(u ^ 0x80000000u)
                             : __uint_as_float(~u);
}
#define ENC_NEG_INF 0x007fffffu   // fenc(-inf)

// ---- generic fill ---------------------------------------------------------
__global__ void fill_u32_kernel(unsigned* __restrict__ p, unsigned v, long n) {
    long i = (long)blockIdx.x * blockDim.x + threadIdx.x;
    if (i < n) p[i] = v;
}

// ---- input projection + ELU: y[n,128] = elu(x[n,:k] @ W[k,128] + b) -------
__global__ void proj_elu_kernel(const float* __restrict__ x,
                                const float* __restrict__ W,
                                const float* __restrict__ b,
                                float* __restrict__ y, int k) {
    __shared__ float xs[16];
    int node = blockIdx.x;
    int j = threadIdx.x;                // 0..127
    if (j < k) xs[j] = x[(long)node * k + j];
    __syncthreads();
    float s = b[j];
    for (int kk = 0; kk < k; ++kk) s += xs[kk] * W[kk * AH + j];
    y[(long)node * AH + j] = s > 0.f ? s : (expf(s) - 1.f);
}

// ---- 128x128 linear via WMMA f32 16x16x4 ----------------------------------
// Block = 256 threads (8 waves).  Block covers 16 rows x 128 cols; wave w
// owns column tile w*16.  K=128 consumed in 32 steps of 4 (one WMMA each).
// A layout (32-bit A, 16x4): lanes 0-15 -> K=k,k+1 ; lanes 16-31 -> K=k+2,k+3.
// C/D layout: VGPR v -> row v (lanes 0-15) / row v+8 (lanes 16-31).
__global__ void linear_wmma_kernel(const float* __restrict__ X,
                                   const float* __restrict__ W,
                                   const float* __restrict__ bias,
                                   float* __restrict__ Y) {
    int wave = threadIdx.x >> 5;        // 0..7 -> column tile
    int lane = threadIdx.x & 31;
    int half = lane >> 4;               // 0 or 1
    int lr   = lane & 15;
    int r0   = blockIdx.x * 16;
    int c0   = wave * 16;

    const float* xrow = X + (long)(r0 + lr) * AH + 2 * half;
    v8f acc = {};
#pragma unroll 4
    for (int k = 0; k < AH; k += 4) {
        v2f a, bm;
        a.x  = xrow[k];
        a.y  = xrow[k + 1];
        bm.x = W[(k + 2 * half) * AH + c0 + lr];
        bm.y = W[(k + 2 * half + 1) * AH + c0 + lr];
        acc = __builtin_amdgcn_wmma_f32_16x16x4_f32(
                  false, a, false, bm, (short)0, acc, false, false);
    }
    float bv = bias[c0 + lr];
    int col = c0 + lr;
#pragma unroll
    for (int v = 0; v < 8; ++v) {
        int row = r0 + v + 8 * half;
        Y[(long)row * AH + col] = acc[v] + bv;
    }
}

// ---- edge pass 1: logits + segment max ------------------------------------
// one wave per (edge, head); lane = feature dim d (0..31)
__global__ void edge_logit_kernel(const int* __restrict__ ei,
                                  const float* __restrict__ ea,
                                  const float* __restrict__ hsrc,
                                  const float* __restrict__ hdst,
                                  const float* __restrict__ Wedge,
                                  const float* __restrict__ att,
                                  float* __restrict__ logit,
                                  unsigned* __restrict__ lmax, int E) {
    long gw = ((long)blockIdx.x * blockDim.x + threadIdx.x) >> 5;
    int lane = threadIdx.x & 31;
    if (gw >= (long)E * NH) return;
    int e = (int)(gw >> 2);
    int h = (int)(gw & 3);
    int src = ei[e];
    int dst = ei[E + e];
    int c = h * HD + lane;

    float ev = ea[e * 4 + 0] * Wedge[0 * AH + c]
             + ea[e * 4 + 1] * Wedge[1 * AH + c]
             + ea[e * 4 + 2] * Wedge[2 * AH + c]
             + ea[e * 4 + 3] * Wedge[3 * AH + c];
    float z = hsrc[(long)src * AH + c] + hdst[(long)dst * AH + c] + ev;
    z = z > 0.f ? z : 0.2f * z;                 // leaky_relu(0.2)
    float p = z * att[c];
#pragma unroll
    for (int off = 16; off > 0; off >>= 1) p += __shfl_xor(p, off, 32);
    if (lane == 0) {
        logit[(long)e * NH + h] = p;
        atomicMax(&lmax[(long)dst * NH + h], fenc(p));
    }
}

// ---- edge pass 2: exp + segment sum ---------------------------------------
__global__ void edge_exp_kernel(const int* __restrict__ ei,
                                float* __restrict__ logit,
                                const unsigned* __restrict__ lmax,
                                float* __restrict__ den, int E) {
    long i = (long)blockIdx.x * blockDim.x + threadIdx.x;
    if (i >= (long)E * NH) return;
    int e = (int)(i >> 2);
    int h = (int)(i & 3);
    int dst = ei[E + e];
    float ex = expf(logit[i] - fdec(lmax[(long)dst * NH + h]));
    logit[i] = ex;                               // reuse buffer for exp values
    atomicAdd(&den[(long)dst * NH + h], ex);
}

// ---- edge pass 3: weighted scatter ----------------------------------------
__global__ void edge_scatter_kernel(const int* __restrict__ ei,
                                    const float* __restrict__ hsrc,
                                    const float* __restrict__ exbuf,
                                    const float* __restrict__ den,
                                    float* __restrict__ accdst, int E) {
    long gw = ((long)blockIdx.x * blockDim.x + threadIdx.x) >> 5;
    int lane = threadIdx.x & 31;
    if (gw >= (long)E * NH) return;
    int e = (int)(gw >> 2);
    int h = (int)(gw & 3);
    int src = ei[e];
    int dst = ei[E + e];
    float alpha = exbuf[(long)e * NH + h] /
                  (den[(long)dst * NH + h] + 1e-16f);
    int c = h * HD + lane;
    atomicAdd(&accdst[(long)dst * AH + c], hsrc[(long)src * AH + c] * alpha);
}

// ---- per-edge-type output bias (broadcast over all dst nodes) -------------
__global__ void add_bias_kernel(float* __restrict__ acc,
                                const float* __restrict__ bias, long n) {
    long i = (long)blockIdx.x * blockDim.x + threadIdx.x;
    if (i < n) acc[i] += bias[i & (AH - 1)];
}

// ---- elementwise ELU ------------------------------------------------------
__global__ void elu_kernel(const float* __restrict__ x, float* __restrict__ y,
                           long n) {
    long i = (long)blockIdx.x * blockDim.x + threadIdx.x;
    if (i < n) {
        float v = x[i];
        y[i] = v > 0.f ? v : (expf(v) - 1.f);
    }
}

// ---- BatchNorm: per-column batch stats ------------------------------------
__global__ void bn_stats_kernel(const float* __restrict__ x, int n,
                                float* __restrict__ mean,
                                float* __restrict__ rstd) {
    int c = blockIdx.x;                 // 0..127
    float s = 0.f, s2 = 0.f;
    for (int r = threadIdx.x; r < n; r += blockDim.x) {
        float v = x[(long)r * AH + c];
        s += v; s2 += v * v;
    }
    __shared__ float sh[256], sh2[256];
    sh[threadIdx.x] = s; sh2[threadIdx.x] = s2;
    __syncthreads();
    for (int off = blockDim.x >> 1; off > 0; off >>= 1) {
        if (threadIdx.x < off) {
            sh[threadIdx.x]  += sh[threadIdx.x + off];
            sh2[threadIdx.x] += sh2[threadIdx.x + off];
        }
        __syncthreads();
    }
    if (threadIdx.x == 0) {
        float m = sh[0] / n;
        float v = sh2[0] / n - m * m;
        mean[c] = m;
        rstd[c] = rsqrtf(v + 1e-5f);
    }
}

__global__ void bn_apply_kernel(float* __restrict__ x,
                                const float* __restrict__ mean,
                                const float* __restrict__ rstd,
                                const float* __restrict__ g,
                                const float* __restrict__ b, long n) {
    long i = (long)blockIdx.x * blockDim.x + threadIdx.x;
    if (i < n) {
        int c = (int)(i & (AH - 1));
        x[i] = (x[i] - mean[c]) * rstd[c] * g[c] + b[c];
    }
}

// ---- output head: one wave per node ---------------------------------------
__global__ void head_kernel(const float* __restrict__ x,
                            const float* __restrict__ W,
                            const float* __restrict__ b,
                            float* __restrict__ out, int n) {
    long gw = ((long)blockIdx.x * blockDim.x + threadIdx.x) >> 5;
    int lane = threadIdx.x & 31;
    if (gw >= n) return;
    float s = 0.f;
#pragma unroll
    for (int c = lane; c < AH; c += 32) s += x[gw * AH + c] * W[c];
#pragma unroll
    for (int off = 16; off > 0; off >>= 1) s += __shfl_xor(s, off, 32);
    if (lane == 0) out[gw] = s + b[0];
}

// ---------------------------------------------------------------------------
extern "C" void kernel_launch(void* const* d_in, const int* in_sizes, int n_in,
                              void* d_out, int out_size, void* d_ws,
                              size_t ws_size, hipStream_t stream) {
    (void)n_in; (void)out_size; (void)ws_size;

    const int NL  = in_sizes[0] / 10;
    const int NR1 = in_sizes[1] / 8;
    const int NR2 = in_sizes[2] / 8;
    const int N[3]   = {NL, NR1, NR2};
    const int KIN[3] = {10, 8, 8};

    struct EdgeT { int ei, ea, s, d; };
    const EdgeT ets[5] = {{3,4,0,0},{5,6,1,1},{7,8,2,2},{9,10,0,1},{11,12,1,0}};

    auto PF = [&](int idx) { return (const float*)d_in[idx]; };
    // params flattened in dict order starting at index 13:
    // proj: 13 + nt*2 (W,b) | per layer L: 19 + L*41 + et*7 ->
    //   Wsrc,bsrc,Wdst,bdst,Wedge,att,bias | bn: 19 + L*41 + 35 + nt*2 (g,b)
    // heads at 101..104.

    // ---- workspace carve-up ----
    float* ws = (float*)d_ws;
    size_t off = 0;
    float* X[3]; float* ACC[3];
    for (int t = 0; t < 3; ++t) { X[t]   = ws + off; off += (size_t)N[t] * AH; }
    for (int t = 0; t < 3; ++t) { ACC[t] = ws + off; off += (size_t)N[t] * AH; }
    float* HSRC = ws + off; off += (size_t)NL * AH;       // max node count
    float* HDST = ws + off; off += (size_t)NL * AH;
    int Emax = in_sizes[3] / 2;                           // ll is largest
    float* LOGIT = ws + off; off += (size_t)Emax * NH;
    unsigned* LMAX = (unsigned*)(ws + off); off += (size_t)NL * NH;
    float* DEN  = ws + off; off += (size_t)NL * NH;
    float* MEAN = ws + off; off += AH;
    float* RSTD = ws + off; off += AH;

    // ---- input projections ----
    for (int t = 0; t < 3; ++t)
        proj_elu_kernel<<<N[t], AH, 0, stream>>>(
            (const float*)d_in[t], PF(13 + t * 2), PF(13 + t * 2 + 1),
            X[t], KIN[t]);

    // ---- layers ----
    for (int L = 0; L < 2; ++L) {
        for (int t = 0; t < 3; ++t) {
            long n = (long)N[t] * AH;
            fill_u32_kernel<<<(int)((n + 255) / 256), 256, 0, stream>>>(
                (unsigned*)ACC[t], 0u, n);
        }
        for (int e = 0; e < 5; ++e) {
            const EdgeT& T = ets[e];
            const int pb = 19 + L * 41 + e * 7;
            const int* ei = (const int*)d_in[T.ei];
            const float* ea = PF(T.ea);
            const int E = in_sizes[T.ei] / 2;

            linear_wmma_kernel<<<N[T.s] / 16, 256, 0, stream>>>(
                X[T.s], PF(pb + 0), PF(pb + 1), HSRC);
            linear_wmma_kernel<<<N[T.d] / 16, 256, 0, stream>>>(
                X[T.d], PF(pb + 2), PF(pb + 3), HDST);

            long nd4 = (long)N[T.d] * NH;
            fill_u32_kernel<<<(int)((nd4 + 255) / 256), 256, 0, stream>>>(
                LMAX, ENC_NEG_INF, nd4);
            fill_u32_kernel<<<(int)((nd4 + 255) / 256), 256, 0, stream>>>(
                (unsigned*)DEN, 0u, nd4);

            long waves = (long)E * NH;                    // one wave each
            edge_logit_kernel<<<(int)((waves + 7) / 8), 256, 0, stream>>>(
                ei, ea, HSRC, HDST, PF(pb + 4), PF(pb + 5), LOGIT, LMAX, E);
            edge_exp_kernel<<<(int)((waves + 255) / 256), 256, 0, stream>>>(
                ei, LOGIT, LMAX, DEN, E);
            edge_scatter_kernel<<<(int)((waves + 7) / 8), 256, 0, stream>>>(
                ei, HSRC, LOGIT, DEN, ACC[T.d], E);

            long ndall = (long)N[T.d] * AH;
            add_bias_kernel<<<(int)((ndall + 255) / 256), 256, 0, stream>>>(
                ACC[T.d], PF(pb + 6), ndall);
        }
        for (int t = 0; t < 3; ++t) {
            const int bnb = 19 + L * 41 + 35 + t * 2;
            long n = (long)N[t] * AH;
            elu_kernel<<<(int)((n + 255) / 256), 256, 0, stream>>>(
                ACC[t], X[t], n);
            bn_stats_kernel<<<AH, 256, 0, stream>>>(X[t], N[t], MEAN, RSTD);
            bn_apply_kernel<<<(int)((n + 255) / 256), 256, 0, stream>>>(
                X[t], MEAN, RSTD, PF(bnb), PF(bnb + 1), n);
        }
    }

    // ---- heads ----
    const int HWI = 19 + 2 * 41;                          // 101
    float* out = (float*)d_out;
    head_kernel<<<(int)(((long)NL * 32 + 255) / 256), 256, 0, stream>>>(
        X[0], PF(HWI), PF(HWI + 1), out, NL);
    head_kernel<<<(int)(((long)NR1 * 32 + 255) / 256), 256, 0, stream>>>(
        X[1], PF(HWI + 2), PF(HWI + 3), out + NL, NR1);
    head_kernel<<<(int)(((long)NR2 * 32 + 255) / 256), 256, 0, stream>>>(
        X[2], PF(HWI + 2), PF(HWI + 3), out + NL + NR1, NR2);
}